// GaussianQuantRegularizer2_6992206758165
// MI455X (gfx1250) — compile-verified
//
#include <hip/hip_runtime.h>
#include <math.h>

// ---------------------------------------------------------------------------
// GaussianQuantRegularizer2 for MI455X (gfx1250, wave32, WMMA).
//
// score[n,k] (argmax target) is a GEMM  A[131072 x 20(pad of 17)] @ B[20 x 1024]
// done with V_WMMA_F32_16X16X4_F32 (fp32 in/out -> same ordering as the fp32
// JAX reference).  B (80KB) and A' (40KB) live in LDS; the 512MB score matrix
// is reduced (max/argmax) on the fly in registers.
// ---------------------------------------------------------------------------

typedef float v8f __attribute__((ext_vector_type(8)));
typedef float v2f __attribute__((ext_vector_type(2)));

#define DIMV 8
#define KCB 1024            // codebook entries
#define KPAD 20             // K-dim 17 padded to 20 -> 5 WMMA steps of K=4
#define ROWS_PER_WG 512
#define THREADS 256         // 8 wave32
#define ROWS_PER_WAVE 64
#define RTILES 4            // 16-row tiles per wave
#define HALF_LOG_2PI 0.9189385332046727f
#define LAMC 1.0f
#define LOG_NC 10.0f
#define TOLC 0.5f
#define N2 131072
#define ZHAT_ELEMS (32 * 8 * 64 * 64)   // 1048576
#define KL_OFF ZHAT_ELEMS
#define IDX_OFF (ZHAT_ELEMS + 1)

// LDS floats: B[1024][20] + A'[20][512] + idx[512] + red[256]
#define LDS_FLOATS (KCB * KPAD + KPAD * ROWS_PER_WG + ROWS_PER_WG + THREADS)

extern "C" __global__ __launch_bounds__(THREADS)
void vq_main_kernel(const float* __restrict__ z,
                    const float* __restrict__ prior,
                    float* __restrict__ out,
                    float* __restrict__ klpart)
{
    extern __shared__ float lds[];
    float* Bl  = lds;                               // [KCB][KPAD]
    float* Al  = Bl + KCB * KPAD;                   // [KPAD][ROWS_PER_WG]
    int*   idxl = (int*)(Al + KPAD * ROWS_PER_WG);  // [ROWS_PER_WG]
    float* red = (float*)(idxl + ROWS_PER_WG);      // [THREADS]

    const int t   = threadIdx.x;
    const int wg  = blockIdx.x;
    const int n0  = wg * ROWS_PER_WG;               // first flat row (b,h,w)
    const int bb  = n0 >> 12;                       // batch (4096 rows / batch)
    const int hw0 = n0 & 4095;

    // ---- Stage B: codebook matrix in LDS, Bl[k][j] -------------------------
    for (int kk = t; kk < KCB; kk += THREADS) {
        float s2 = 0.f;
        #pragma unroll
        for (int d = 0; d < DIMV; ++d) {
            float s = prior[kk * DIMV + d];
            Bl[kk * KPAD + d]     = -0.5f * s * s;  // pairs with inv_var
            Bl[kk * KPAD + 8 + d] = s;              // pairs with mu*inv_var
            s2 += s * s;
        }
        // -BETA*lp[k] with BETA=1
        Bl[kk * KPAD + 16] = 0.5f * s2 + 8.0f * HALF_LOG_2PI;
        Bl[kk * KPAD + 17] = 0.f;
        Bl[kk * KPAD + 18] = 0.f;
        Bl[kk * KPAD + 19] = 0.f;
    }

    // ---- Stage A': per-row features in LDS + faithful KL loss --------------
    float klacc = 0.f;
    #pragma unroll
    for (int rep = 0; rep < 2; ++rep) {
        int row   = rep * THREADS + t;              // 0..511
        int zbase = bb * 65536 + hw0 + row;         // + c*4096 per channel
        float kl = 0.f;
        #pragma unroll
        for (int d = 0; d < DIMV; ++d) {
            float mu = z[zbase + d * 4096];
            float lv = z[zbase + (d + 8) * 4096];
            lv = fminf(fmaxf(lv, -30.f), 20.f);
            float var = expf(lv);
            float iv  = 1.0f / var;                 // 1/std^2
            Al[d * ROWS_PER_WG + row]       = iv;
            Al[(8 + d) * ROWS_PER_WG + row] = mu * iv;
            kl += mu * mu + var - 1.f - lv;
        }
        Al[16 * ROWS_PER_WG + row] = 1.f;           // picks up -beta*lp row
        Al[17 * ROWS_PER_WG + row] = 0.f;
        Al[18 * ROWS_PER_WG + row] = 0.f;
        Al[19 * ROWS_PER_WG + row] = 0.f;
        float kl2 = 0.7213f * kl;                   // 1.4426 * 0.5
        float w = (kl2 > LOG_NC + TOLC) ? LAMC
                : ((kl2 >= LOG_NC - TOLC) ? 1.0f : LAMC);
        klacc += w * kl2;
    }
    __syncthreads();

    // ---- WMMA GEMM + streaming argmax --------------------------------------
    const int lane = t & 31;
    const int wv   = t >> 5;           // wave 0..7
    const int half = lane >> 4;        // 0: K even pair, 1: K odd pair (layout)
    const int r    = lane & 15;        // row-in-tile / col-in-tile

    // A operands: loaded once, reused for all 64 column tiles.
    // f32 16x4 A layout: lanes 0-15 hold M, VGPR0=K0(K2 for hi lanes), VGPR1=K1(K3).
    v2f Aop[RTILES][5];
    #pragma unroll
    for (int rt = 0; rt < RTILES; ++rt) {
        int row = wv * ROWS_PER_WAVE + rt * 16 + r;
        #pragma unroll
        for (int s = 0; s < 5; ++s) {
            int j = 4 * s + 2 * half;
            v2f a;
            a.x = Al[j * ROWS_PER_WG + row];
            a.y = Al[(j + 1) * ROWS_PER_WG + row];
            Aop[rt][s] = a;
        }
    }

    float bestv[RTILES][8];
    int   besti[RTILES][8];
    #pragma unroll
    for (int rt = 0; rt < RTILES; ++rt)
        #pragma unroll
        for (int i = 0; i < 8; ++i) { bestv[rt][i] = -3.4e38f; besti[rt][i] = 0; }

    const v8f zero8 = {0.f, 0.f, 0.f, 0.f, 0.f, 0.f, 0.f, 0.f};

    for (int tile = 0; tile < 64; ++tile) {
        int col = tile * 16 + r;
        // B operand: 4x16 f32, rows K0/K2 in VGPR0, K1/K3 in VGPR1 -> one b64/step
        v2f Bop[5];
        #pragma unroll
        for (int s = 0; s < 5; ++s) {
            int j = 4 * s + 2 * half;
            Bop[s] = *(const v2f*)&Bl[col * KPAD + j];   // 8B-aligned ds_load_b64
        }
        v8f acc[RTILES];
        #pragma unroll
        for (int rt = 0; rt < RTILES; ++rt) acc[rt] = zero8;
        #pragma unroll
        for (int s = 0; s < 5; ++s)
            #pragma unroll
            for (int rt = 0; rt < RTILES; ++rt)
                acc[rt] = __builtin_amdgcn_wmma_f32_16x16x4_f32(
                    false, Aop[rt][s], false, Bop[s],
                    (short)0, acc[rt], false, false);
        // streaming per-lane argmax (ascending tiles + strict '>' == first max)
        #pragma unroll
        for (int rt = 0; rt < RTILES; ++rt)
            #pragma unroll
            for (int i = 0; i < 8; ++i) {
                float v = acc[rt][i];
                if (v > bestv[rt][i]) { bestv[rt][i] = v; besti[rt][i] = col; }
            }
    }

    // ---- cross-lane reduction (16 lanes per row group), low-idx tie-break --
    #pragma unroll
    for (int rt = 0; rt < RTILES; ++rt)
        #pragma unroll
        for (int i = 0; i < 8; ++i) {
            float v = bestv[rt][i];
            int   id = besti[rt][i];
            #pragma unroll
            for (int off = 8; off >= 1; off >>= 1) {
                float ov = __shfl_xor(v, off, 32);
                int   oi = __shfl_xor(id, off, 32);
                if (ov > v || (ov == v && oi < id)) { v = ov; id = oi; }
            }
            if (r == 0) {
                // lanes 0-15: rows M=i ; lanes 16-31: rows M=i+8
                int row = wv * ROWS_PER_WAVE + rt * 16 + half * 8 + i;
                idxl[row] = id;
            }
        }
    __syncthreads();

    // ---- outputs ------------------------------------------------------------
    // zhat: [32, 8, 64, 64]; coalesced per channel.
    #pragma unroll
    for (int d = 0; d < DIMV; ++d)
        #pragma unroll
        for (int rep = 0; rep < 2; ++rep) {
            int row = rep * THREADS + t;
            out[bb * 32768 + d * 4096 + hw0 + row] = prior[idxl[row] * DIMV + d];
        }
    // indices: [32, 1, 64, 64] stored as float
    #pragma unroll
    for (int rep = 0; rep < 2; ++rep) {
        int row = rep * THREADS + t;
        out[IDX_OFF + n0 + row] = (float)idxl[row];
    }
    // deterministic KL partial: fixed-order LDS tree
    red[t] = klacc;
    __syncthreads();
    for (int s = 128; s > 0; s >>= 1) {
        if (t < s) red[t] += red[t + s];
        __syncthreads();
    }
    if (t == 0) klpart[wg] = red[0];
}

extern "C" __global__ __launch_bounds__(256)
void vq_kl_final_kernel(const float* __restrict__ klpart, float* __restrict__ out)
{
    __shared__ float red[256];
    int t = threadIdx.x;
    red[t] = klpart[t];   // 256 workgroup partials
    __syncthreads();
    for (int s = 128; s > 0; s >>= 1) {
        if (t < s) red[t] += red[t + s];
        __syncthreads();
    }
    if (t == 0) out[KL_OFF] = red[0] * (LAMC / (float)N2);
}

extern "C" void kernel_launch(void* const* d_in, const int* in_sizes, int n_in,
                              void* d_out, int out_size, void* d_ws, size_t ws_size,
                              hipStream_t stream)
{
    const float* z     = (const float*)d_in[0];
    const float* noise = (const float*)d_in[1];   // cancels in straight-through
    const float* prior = (const float*)d_in[2];
    (void)noise; (void)in_sizes; (void)n_in; (void)out_size;

    float* out    = (float*)d_out;
    float* klpart = (float*)d_ws;                 // 256 floats of scratch
    (void)ws_size;

    const size_t lds_bytes = (size_t)LDS_FLOATS * sizeof(float);  // ~123 KB
    hipFuncSetAttribute((const void*)vq_main_kernel,
                        hipFuncAttributeMaxDynamicSharedMemorySize,
                        (int)lds_bytes);

    vq_main_kernel<<<N2 / ROWS_PER_WG, THREADS, lds_bytes, stream>>>(
        z, prior, out, klpart);
    vq_kl_final_kernel<<<1, 256, 0, stream>>>(klpart, out);
}